// MipNeRFRenderer_11587821765202
// MI455X (gfx1250) — compile-verified
//
#include <hip/hip_runtime.h>

// MipNeRF volumetric render + hierarchical sampling for MI455X (gfx1250).
// Bandwidth-bound: ~1.08 GB total traffic => ~46us floor @ 23.3 TB/s.
// Transmittance (exclusive cumprod) computed in log space as a matrix-unit
// prefix scan: Y_c = X_c * U (U = upper-triangular ones) via chained
// V_WMMA_F32_16X16X4_F32; 16 rays per wave, 8 chunks of 16, inter-chunk
// carries via lane-15 broadcast shuffles. Input staging uses
// GLOBAL_LOAD_ASYNC_TO_LDS_B128 (ASYNCcnt) when available.

typedef float v2f __attribute__((ext_vector_type(2)));
typedef float v8f __attribute__((ext_vector_type(8)));
typedef int   v4i __attribute__((vector_size(16)));   // matches builtin param type

#define GAS __attribute__((address_space(1)))
#define LAS __attribute__((address_space(3)))

#define R_TOT 262144
#define NCNT 128
#define NFC 128
#define WAVES 4
#define RAYS_PER_WAVE 16
#define RAYS_PER_BLOCK (WAVES * RAYS_PER_WAVE)
#define RS 132   // padded LDS row stride (floats); %4==0 for b128, %64==4 -> no bank conflicts

#if __has_builtin(__builtin_amdgcn_global_load_async_to_lds_b128)
#define USE_ASYNC_LDS 1
#else
#define USE_ASYNC_LDS 0
#endif

__global__ __launch_bounds__(128) void nerf_render_sample(
    const float* __restrict__ dens,   // [R,128,1]
    const float* __restrict__ cols,   // [R,128,3]
    const float* __restrict__ tv,     // [R,128]
    const float* __restrict__ uu,     // [R,128]
    float* __restrict__ out)          // rgb(3R) | depth(R) | acc(R) | w(128R) | smp(128R)
{
    const int tid  = threadIdx.x;
    const int lane = tid & 31;
    const int wid  = tid >> 5;
    const int n    = lane & 15;   // column (N) for B/C/D, row (M) for A
    const int h    = lane >> 4;   // half-wave select
    const int rayBase = blockIdx.x * RAYS_PER_BLOCK + wid * RAYS_PER_WAVE;

    extern __shared__ float smem[];
    float* tS = smem + (size_t)wid * (16 * RS);                 // t_vals slab
    float* dS = smem + (size_t)(WAVES + wid) * (16 * RS);       // densities, reused as cdf
    float* wS = smem + (size_t)(2 * WAVES + wid) * (16 * RS);   // weights (+acc at col 130)

    // ---- prefetch colors for this block (global_prefetch_b8 path) ----
    {
        const float* cBlk = cols + (size_t)(blockIdx.x) * RAYS_PER_BLOCK * NCNT * 3;
        #pragma unroll
        for (int i = 0; i < 6; ++i)
            __builtin_prefetch(cBlk + ((size_t)tid + (size_t)i * 128) * 32, 0, 0);
    }

    // ---- staging: t_vals & densities -> LDS ----
    {
        const float* tG = tv   + (size_t)rayBase * NCNT;
        const float* dG = dens + (size_t)rayBase * NCNT;
#if USE_ASYNC_LDS
        // Direct global->LDS DMA (no VGPR round-trip), tracked by ASYNCcnt.
        float* tGm = const_cast<float*>(tG);
        float* dGm = const_cast<float*>(dG);
        #pragma unroll
        for (int i = 0; i < 16; ++i) {
            int idx = i * 32 + lane;             // 0..511 float4 slots (16 rays x 128)
            int el = idx * 4, row = el >> 7, col = el & 127;
            __builtin_amdgcn_global_load_async_to_lds_b128(
                (GAS v4i*)(tGm + el), (LAS v4i*)&tS[row * RS + col], 0, 0);
            __builtin_amdgcn_global_load_async_to_lds_b128(
                (GAS v4i*)(dGm + el), (LAS v4i*)&dS[row * RS + col], 0, 0);
        }
        asm volatile("s_wait_asynccnt 0" ::: "memory");
#else
        #pragma unroll
        for (int i = 0; i < 16; ++i) {
            int idx = i * 32 + lane;
            float4 vt = ((const float4*)tG)[idx];
            float4 vd = ((const float4*)dG)[idx];
            int el = idx * 4, row = el >> 7, col = el & 127;
            *(float4*)&tS[row * RS + col] = vt;
            *(float4*)&dS[row * RS + col] = vd;
        }
#endif
    }
    __syncthreads();

    // ---- B operands: U[k][n] = (k <= n), split into 4 K-slices of 4 ----
    v2f B[4];
    #pragma unroll
    for (int q = 0; q < 4; ++q) {
        B[q].x = ((4 * q + 2 * h)     <= n) ? 1.0f : 0.0f;
        B[q].y = ((4 * q + 2 * h + 1) <= n) ? 1.0f : 0.0f;
    }

    float o[8], eo[8], accP[8], depP[8], rgbP[8][3];
    #pragma unroll
    for (int j = 0; j < 8; ++j) {
        o[j] = 0.0f; eo[j] = 1.0f; accP[j] = 0.0f; depP[j] = 0.0f;
        rgbP[j][0] = rgbP[j][1] = rgbP[j][2] = 0.0f;
    }
    const int src15 = (lane & 16) | 15;

    // ---- 8 chunks of 16 samples: WMMA prefix-scan + weights + reductions ----
    #pragma unroll 2
    for (int c = 0; c < 8; ++c) {
        v8f Y = {};
        #pragma unroll
        for (int q = 0; q < 4; ++q) {
            int e0 = c * 16 + q * 4 + 2 * h;           // A: lane n = ray row M
            float t0 = tS[n * RS + e0];
            float t1 = tS[n * RS + e0 + 1];
            float t2 = tS[n * RS + e0 + 2];            // pad-safe at e0==126
            float s0 = fmaxf(dS[n * RS + e0], 0.0f);
            float s1 = fmaxf(dS[n * RS + e0 + 1], 0.0f);
            float dist0 = t1 - t0;
            float dist1 = (e0 + 1 == 127) ? 1e10f : (t2 - t1);
            v2f a;
            a.x = __logf(__expf(-s0 * dist0) + 1e-10f);    // log(1-alpha+1e-10)
            a.y = __logf(__expf(-s1 * dist1) + 1e-10f);
            Y = __builtin_amdgcn_wmma_f32_16x16x4_f32(
                    false, a, false, B[q], (short)0, Y, false, false);
        }
        // C/D layout: lane -> column n, VGPR j -> ray row (8h + j)
        #pragma unroll
        for (int j = 0; j < 8; ++j) {
            float sIncl = Y[j] + o[j];
            float eIncl = __expf(sIncl);
            float eExcl = __shfl_up(eIncl, 1, 32);
            if (n == 0) eExcl = eo[j];
            float w = fmaf(1e-10f, eExcl, eExcl - eIncl);   // alpha * trans
            o[j]  = __shfl(sIncl, src15, 32);               // carry to next chunk
            eo[j] = __shfl(eIncl, src15, 32);
            int row = 8 * h + j;
            int k   = c * 16 + n;
            wS[row * RS + k] = w;
            float tmn = tS[row * RS + k];
            accP[j] += w;
            depP[j]  = fmaf(w, tmn, depP[j]);
            const float* cp = cols + ((size_t)(rayBase + row) * NCNT + k) * 3;
            rgbP[j][0] = fmaf(w, cp[0], rgbP[j][0]);
            rgbP[j][1] = fmaf(w, cp[1], rgbP[j][1]);
            rgbP[j][2] = fmaf(w, cp[2], rgbP[j][2]);
        }
    }

    // ---- cross-lane reduction over n (each half-wave independently) ----
    #pragma unroll
    for (int off = 1; off < 16; off <<= 1) {
        #pragma unroll
        for (int j = 0; j < 8; ++j) {
            accP[j] += __shfl_xor(accP[j], off, 32);
            depP[j] += __shfl_xor(depP[j], off, 32);
            rgbP[j][0] += __shfl_xor(rgbP[j][0], off, 32);
            rgbP[j][1] += __shfl_xor(rgbP[j][1], off, 32);
            rgbP[j][2] += __shfl_xor(rgbP[j][2], off, 32);
        }
    }
    if (n == 0) {
        #pragma unroll
        for (int j = 0; j < 8; ++j) {
            int ray = rayBase + 8 * h + j;
            out[(size_t)ray * 3 + 0] = rgbP[j][0];
            out[(size_t)ray * 3 + 1] = rgbP[j][1];
            out[(size_t)ray * 3 + 2] = rgbP[j][2];
            out[(size_t)3 * R_TOT + ray] = depP[j];
            out[(size_t)4 * R_TOT + ray] = accP[j];
            wS[(8 * h + j) * RS + 130] = accP[j];   // stash for CDF normalizer
        }
    }
    __syncthreads();

    // ---- per-ray CDF over interior weights (reuse density slab) ----
    if (lane < 16) {
        int r = lane;
        float w0   = wS[r * RS + 0];
        float w127 = wS[r * RS + 127];
        float acc  = wS[r * RS + 130];
        float Z = (acc - w0 - w127) + 126.0f * 1e-5f;
        float invZ = 1.0f / Z;
        dS[r * RS + 0] = 0.0f;
        float run = 0.0f;
        for (int j = 1; j <= 126; ++j) {
            run += wS[r * RS + j] + 1e-5f;
            dS[r * RS + j] = run * invZ;
        }
    }
    __syncthreads();

    // ---- inverse-CDF sampling: linearized item index => coalesced u/samples ----
    {
        const float* uW = uu + (size_t)rayBase * NFC;
        float* smpW = out + (size_t)133 * R_TOT + (size_t)rayBase * NFC;
        for (int it = 0; it < 64; ++it) {
            int item = it * 32 + lane;
            int r = item >> 7;
            float uval = uW[item];
            const float* cdf = dS + r * RS;
            int lo = 0, cnt = 127;                  // upper_bound over cdf[0..126]
            while (cnt > 0) {
                int half = cnt >> 1;
                int mid = lo + half;
                if (cdf[mid] <= uval) { lo = mid + 1; cnt -= half + 1; }
                else cnt = half;
            }
            int below = lo - 1; below = (below < 0) ? 0 : below; below = (below > 126) ? 126 : below;
            int above = (lo > 126) ? 126 : lo;
            float cb = cdf[below], ca = cdf[above];
            float bb = tS[r * RS + below], ba = tS[r * RS + above];
            float den = ca - cb;
            den = (den < 1e-5f) ? 1.0f : den;
            float tt = (uval - cb) / den;
            smpW[item] = fmaf(tt, ba - bb, bb);
        }
    }

    // ---- weights: LDS -> global, coalesced float4 ----
    {
        float* wOut = out + (size_t)5 * R_TOT + (size_t)rayBase * NCNT;
        #pragma unroll
        for (int it = 0; it < 16; ++it) {
            int idx4 = it * 32 + lane;
            int row = idx4 >> 5;
            int col = (idx4 & 31) * 4;
            float4 v = *(const float4*)&wS[row * RS + col];
            ((float4*)wOut)[idx4] = v;
        }
    }
}

extern "C" void kernel_launch(void* const* d_in, const int* in_sizes, int n_in,
                              void* d_out, int out_size, void* d_ws, size_t ws_size,
                              hipStream_t stream) {
    (void)in_sizes; (void)n_in; (void)out_size; (void)d_ws; (void)ws_size;
    const float* dens = (const float*)d_in[0];
    const float* cols = (const float*)d_in[1];
    const float* tv   = (const float*)d_in[2];
    const float* u    = (const float*)d_in[3];
    float* out = (float*)d_out;

    dim3 grid(R_TOT / RAYS_PER_BLOCK);   // 4096
    dim3 block(128);                     // 4 waves, 16 rays/wave
    size_t lds = (size_t)3 * WAVES * 16 * RS * sizeof(float);  // ~99 KB
    nerf_render_sample<<<grid, block, lds, stream>>>(dens, cols, tv, u, out);
}